// GaussianPointCloudRasterisation_60782377173331
// MI455X (gfx1250) — compile-verified
//
#include <hip/hip_runtime.h>
#include <math.h>

#define GS_TILE 16
#define GS_NEAR 0.8f
#define GS_FAR 1000.0f
#define GS_ALPHA_THR (1.0f / 255.0f)
#define GS_T_THR 0.0001f

typedef __attribute__((ext_vector_type(2))) float v2f;
typedef __attribute__((ext_vector_type(8))) float v8f;

// D(16x16) = A(16x4, M=gaussian x K=feature) * B(4x16, K x N=pixel) + C
static __device__ __forceinline__ v8f wmma_gs(v2f a, v2f b, v8f c) {
  return __builtin_amdgcn_wmma_f32_16x16x4_f32(
      /*neg_a=*/false, a, /*neg_b=*/false, b,
      /*c_mod=*/(short)0, c, /*reuse_a=*/false, /*reuse_b=*/false);
}

// Workspace SoA layout (float slots of size N each):
// pre:    [0]=key [1..6]=g0..g5 [7]=alpha [8]=z [9..11]=rgb [12]=bnd(int) [13]=valid(int)
// sorted: same indices at base 14*N (slot 0 unused)
#define PRE(ws, a, N) ((ws) + (size_t)(a) * (N))
#define SRT(ws, a, N) ((ws) + (size_t)(14 + (a)) * (N))

static __device__ __forceinline__ int clampi(int v, int lo, int hi) {
  return v < lo ? lo : (v > hi ? hi : v);
}

__global__ __launch_bounds__(256) void gs_preprocess(
    const float* __restrict__ pc, const float* __restrict__ feat,
    const int* __restrict__ inval, const float* __restrict__ Km,
    const float* __restrict__ qv, const float* __restrict__ tv,
    const int* __restrict__ Hp, const int* __restrict__ Wp,
    const int* __restrict__ bandp, float* __restrict__ ws, int N) {
  int i = blockIdx.x * blockDim.x + threadIdx.x;
  if (i >= N) return;

  // ---- camera pose (uniform; recompute per thread, ~50 flops) ----
  float qx = qv[0], qy = qv[1], qz = qv[2], qw = qv[3];
  float qn = sqrtf(qx * qx + qy * qy + qz * qz + qw * qw);
  qx /= qn; qy /= qn; qz /= qn; qw /= qn;
  float Rwc[3][3];
  Rwc[0][0] = 1.0f - 2.0f * (qy * qy + qz * qz);
  Rwc[0][1] = 2.0f * (qx * qy - qw * qz);
  Rwc[0][2] = 2.0f * (qx * qz + qw * qy);
  Rwc[1][0] = 2.0f * (qx * qy + qw * qz);
  Rwc[1][1] = 1.0f - 2.0f * (qx * qx + qz * qz);
  Rwc[1][2] = 2.0f * (qy * qz - qw * qx);
  Rwc[2][0] = 2.0f * (qx * qz - qw * qy);
  Rwc[2][1] = 2.0f * (qy * qz + qw * qx);
  Rwc[2][2] = 1.0f - 2.0f * (qx * qx + qy * qy);
  float Rcw[3][3];
#pragma unroll
  for (int r = 0; r < 3; ++r)
#pragma unroll
    for (int c = 0; c < 3; ++c) Rcw[r][c] = Rwc[c][r];
  float twx = tv[0], twy = tv[1], twz = tv[2];
  float tc[3];
#pragma unroll
  for (int r = 0; r < 3; ++r)
    tc[r] = -(Rcw[r][0] * twx + Rcw[r][1] * twy + Rcw[r][2] * twz);

  float fx = Km[0], fy = Km[4], cx = Km[2], cy = Km[5];
  int H = *Hp, W = *Wp;

  // ---- project ----
  float px = pc[3 * i + 0], py = pc[3 * i + 1], pz = pc[3 * i + 2];
  float xc = Rcw[0][0] * px + Rcw[0][1] * py + Rcw[0][2] * pz + tc[0];
  float yc = Rcw[1][0] * px + Rcw[1][1] * py + Rcw[1][2] * pz + tc[1];
  float zc = Rcw[2][0] * px + Rcw[2][1] * py + Rcw[2][2] * pz + tc[2];
  float invz = 1.0f / zc;
  float u = fx * xc * invz + cx;
  float v = fy * yc * invz + cy;
  int valid = (inval[i] == 0) && (zc > GS_NEAR) && (zc < GS_FAR) &&
              (u > -4.0f * GS_TILE) && (u < (float)W + 4.0f * GS_TILE) &&
              (v > -4.0f * GS_TILE) && (v < (float)H + 4.0f * GS_TILE);

  // ---- 3D covariance -> 2D conic ----
  const float* f = feat + 56 * (size_t)i;
  float ax = f[0], ay = f[1], az = f[2], aw = f[3];
  float an = sqrtf(ax * ax + ay * ay + az * az + aw * aw);
  ax /= an; ay /= an; az /= an; aw /= an;
  float Rp[3][3];
  Rp[0][0] = 1.0f - 2.0f * (ay * ay + az * az);
  Rp[0][1] = 2.0f * (ax * ay - aw * az);
  Rp[0][2] = 2.0f * (ax * az + aw * ay);
  Rp[1][0] = 2.0f * (ax * ay + aw * az);
  Rp[1][1] = 1.0f - 2.0f * (ax * ax + az * az);
  Rp[1][2] = 2.0f * (ay * az - aw * ax);
  Rp[2][0] = 2.0f * (ax * az - aw * ay);
  Rp[2][1] = 2.0f * (ay * az + aw * ax);
  Rp[2][2] = 1.0f - 2.0f * (ax * ax + ay * ay);
  float s0 = expf(f[4]), s1 = expf(f[5]), s2 = expf(f[6]);
  // L = Rcw * (Rp * diag(s));  Sigma_cam = L * L^T
  float L[3][3];
#pragma unroll
  for (int r = 0; r < 3; ++r) {
    L[r][0] = (Rcw[r][0] * Rp[0][0] + Rcw[r][1] * Rp[1][0] + Rcw[r][2] * Rp[2][0]) * s0;
    L[r][1] = (Rcw[r][0] * Rp[0][1] + Rcw[r][1] * Rp[1][1] + Rcw[r][2] * Rp[2][1]) * s1;
    L[r][2] = (Rcw[r][0] * Rp[0][2] + Rcw[r][1] * Rp[1][2] + Rcw[r][2] * Rp[2][2]) * s2;
  }
  float S00 = L[0][0] * L[0][0] + L[0][1] * L[0][1] + L[0][2] * L[0][2];
  float S01 = L[0][0] * L[1][0] + L[0][1] * L[1][1] + L[0][2] * L[1][2];
  float S02 = L[0][0] * L[2][0] + L[0][1] * L[2][1] + L[0][2] * L[2][2];
  float S11 = L[1][0] * L[1][0] + L[1][1] * L[1][1] + L[1][2] * L[1][2];
  float S12 = L[1][0] * L[2][0] + L[1][1] * L[2][1] + L[1][2] * L[2][2];
  float S22 = L[2][0] * L[2][0] + L[2][1] * L[2][1] + L[2][2] * L[2][2];
  float j00 = fx * invz, j02 = -fx * xc * invz * invz;
  float j11 = fy * invz, j12 = -fy * yc * invz * invz;
  float m00 = j00 * S00 + j02 * S02;
  float m01 = j00 * S01 + j02 * S12;
  float m02 = j00 * S02 + j02 * S22;
  float n11 = j11 * S11 + j12 * S12;
  float n12 = j11 * S12 + j12 * S22;
  float cov00 = m00 * j00 + m02 * j02;
  float cov01 = m01 * j11 + m02 * j12;
  float cov11 = n11 * j11 + n12 * j12;
  float A = cov00 + 0.3f, Bc = cov01, C = cov11 + 0.3f;
  float det = fmaxf(A * C - Bc * Bc, 1e-9f);
  float ca = C / det, cb = -Bc / det, cc = A / det;
  float mid = 0.5f * (A + C);
  float lam = mid + sqrtf(fmaxf(mid * mid - det, 1e-9f));
  float radii = ceilf(3.0f * sqrtf(lam));
  float alpha = 1.0f / (1.0f + expf(-f[7]));

  // ---- SH color ----
  float dx = px - twx, dy = py - twy, dz = pz - twz;
  float dn = sqrtf(dx * dx + dy * dy + dz * dz);
  dx /= dn; dy /= dn; dz /= dn;
  float xx = dx * dx, yy = dy * dy, zz = dz * dz;
  float xy = dx * dy, yz = dy * dz, xz = dx * dz;
  float bs[16];
  bs[0] = 0.28209479177387814f;
  bs[1] = -0.4886025119029199f * dy;
  bs[2] = 0.4886025119029199f * dz;
  bs[3] = -0.4886025119029199f * dx;
  bs[4] = 1.0925484305920792f * xy;
  bs[5] = -1.0925484305920792f * yz;
  bs[6] = 0.31539156525252005f * (2.0f * zz - xx - yy);
  bs[7] = -1.0925484305920792f * xz;
  bs[8] = 0.5462742152960396f * (xx - yy);
  bs[9] = -0.5900435899266435f * dy * (3.0f * xx - yy);
  bs[10] = 2.890611442640554f * xy * dz;
  bs[11] = -0.4570457994644658f * dy * (4.0f * zz - xx - yy);
  bs[12] = 0.3731763325901154f * dz * (2.0f * zz - 3.0f * xx - 3.0f * yy);
  bs[13] = -0.4570457994644658f * dx * (4.0f * zz - xx - yy);
  bs[14] = 1.445305721320277f * dz * (xx - yy);
  bs[15] = -0.5900435899266435f * dx * (xx - 3.0f * yy);
  int band = *bandp;
  int KK = (band + 1) * (band + 1);
  if (KK > 16) KK = 16;
  float col[3];
#pragma unroll
  for (int c = 0; c < 3; ++c) {
    float acc = 0.0f;
    for (int k = 0; k < KK; ++k) acc += bs[k] * f[8 + c * 16 + k];
    col[c] = fmaxf(acc + 0.5f, 0.0f);
  }

  // ---- tile bounds (packed) ----
  int tilesX = W / GS_TILE, tilesY = H / GS_TILE;
  int bnd = 1;  // empty: t0x=1, t1x=0
  if (valid) {
    int t0x = clampi((int)floorf((u - radii) / (float)GS_TILE), 0, tilesX - 1);
    int t1x = clampi((int)floorf((u + radii) / (float)GS_TILE), 0, tilesX - 1);
    int t0y = clampi((int)floorf((v - radii) / (float)GS_TILE), 0, tilesY - 1);
    int t1y = clampi((int)floorf((v + radii) / (float)GS_TILE), 0, tilesY - 1);
    bnd = (t0x & 255) | ((t1x & 255) << 8) | ((t0y & 255) << 16) | ((t1y & 255) << 24);
  }

  // ---- WMMA gaussian features: power = g0*px^2+g1*py^2+g2*px*py+g3*px+g4*py+g5 ----
  float g0 = -0.5f * ca;
  float g1 = -0.5f * cc;
  float g2 = -cb;
  float g3 = ca * u + cb * v;
  float g4 = cc * v + cb * u;
  float g5 = -0.5f * (ca * u * u + cc * v * v) - cb * u * v;

  PRE(ws, 0, N)[i] = valid ? zc : __int_as_float(0x7f800000);  // key (+inf invalid)
  PRE(ws, 1, N)[i] = g0;
  PRE(ws, 2, N)[i] = g1;
  PRE(ws, 3, N)[i] = g2;
  PRE(ws, 4, N)[i] = g3;
  PRE(ws, 5, N)[i] = g4;
  PRE(ws, 6, N)[i] = g5;
  PRE(ws, 7, N)[i] = alpha;
  PRE(ws, 8, N)[i] = zc;
  PRE(ws, 9, N)[i] = col[0];
  PRE(ws, 10, N)[i] = col[1];
  PRE(ws, 11, N)[i] = col[2];
  ((int*)PRE(ws, 12, N))[i] = bnd;
  ((int*)PRE(ws, 13, N))[i] = valid;
}

// Stable O(N^2) rank sort + scatter (N=2048 -> 4M cmps, trivial, fully deterministic)
__global__ __launch_bounds__(256) void gs_sort_scatter(float* __restrict__ ws, int N) {
  int i = blockIdx.x * blockDim.x + threadIdx.x;
  if (i >= N) return;
  const float* key = PRE(ws, 0, N);
  float ki = key[i];
  int rank = 0;
  for (int j = 0; j < N; ++j) {
    float kj = key[j];
    rank += (kj < ki) || ((kj == ki) && (j < i));
  }
#pragma unroll
  for (int a = 1; a <= 11; ++a) SRT(ws, a, N)[rank] = PRE(ws, a, N)[i];
  ((int*)SRT(ws, 12, N))[rank] = ((const int*)PRE(ws, 12, N))[i];
  ((int*)SRT(ws, 13, N))[rank] = ((const int*)PRE(ws, 13, N))[i];
}

__global__ __launch_bounds__(256) void gs_raster(
    const float* __restrict__ ws, const int* __restrict__ Hp,
    const int* __restrict__ Wp, float* __restrict__ out, int N) {
  __shared__ float sPow[16 * 256];  // 16 gaussians x 256 pixels, 16 KB
  __shared__ float sAl[16], sZ[16], sR[16], sG[16], sB[16];

  int H = *Hp, W = *Wp;
  int tilesX = W / GS_TILE, tilesY = H / GS_TILE;
  int nTiles = tilesX * tilesY;
  int HW = H * W;

  const float* Sg0 = SRT(ws, 1, N);
  const float* Sg1 = SRT(ws, 2, N);
  const float* Sg2 = SRT(ws, 3, N);
  const float* Sg3 = SRT(ws, 4, N);
  const float* Sg4 = SRT(ws, 5, N);
  const float* Sg5 = SRT(ws, 6, N);
  const float* Sal = SRT(ws, 7, N);
  const float* Sz = SRT(ws, 8, N);
  const float* Sr = SRT(ws, 9, N);
  const float* Sgc = SRT(ws, 10, N);
  const float* Sb = SRT(ws, 11, N);
  const int* Sbnd = (const int*)SRT(ws, 12, N);
  const int* Svld = (const int*)SRT(ws, 13, N);

  int t = threadIdx.x;
  int lane = t & 31;
  int wv = t >> 5;
  int col = lane & 15;
  int half = lane >> 4;
  int px_i = t & 15, py_i = t >> 4;
  int nb = (N + 15) / 16;

  // A-feature pointers, selected per lane-half (v_cndmask on address, no exec games)
  const float* pA1x = half ? Sg2 : Sg0;
  const float* pA1y = half ? Sg3 : Sg1;

  // sPow store bases (tile-invariant): M = vi + 8*half, pixel = row*16 + col
  int pbase0 = (2 * wv + 0) * 16 + col + (half ? 8 * 256 : 0);
  int pbase1 = (2 * wv + 1) * 16 + col + (half ? 8 * 256 : 0);

  for (int tile = blockIdx.x; tile < nTiles; tile += gridDim.x) {
    int ty = tile / tilesX, tx = tile % tilesX;

    // ---- hoisted pixel-feature (B) operands: depend only on tile & lane ----
    // K layout: lanes 0-15 hold K0,K1; lanes 16-31 hold K2,K3.
    // WMMA1: F = [px^2, py^2, px*py, px]; WMMA2: F = [py, 1, 0, 0]
    float Bx = (float)(tx * GS_TILE + col) + 0.5f;
    float By0 = (float)(ty * GS_TILE + 2 * wv + 0) + 0.5f;
    float By1 = (float)(ty * GS_TILE + 2 * wv + 1) + 0.5f;
    v2f B1g0 = {half ? Bx * By0 : Bx * Bx, half ? Bx : By0 * By0};
    v2f B1g1 = {half ? Bx * By1 : Bx * Bx, half ? Bx : By1 * By1};
    v2f B2g0 = {half ? 0.0f : By0, half ? 0.0f : 1.0f};
    v2f B2g1 = {half ? 0.0f : By1, half ? 0.0f : 1.0f};

    float T = 1.0f, rgb0 = 0.0f, rgb1 = 0.0f, rgb2 = 0.0f;
    float dep = 0.0f, acc = 0.0f;
    int cnt = 0;

    for (int b = 0; b < nb; ++b) {
      // barrier #1: uniform early-out + guards LDS reuse from previous batch
      int alive = __syncthreads_count(T > GS_T_THR);
      if (alive == 0) break;

      // ---- stage per-gaussian scalars (wave 0, lanes 0-15) ----
      if (t < 16) {
        int gi = b * 16 + t;
        float al = 0.0f, z = 0.0f, r = 0.0f, g = 0.0f, bl = 0.0f;
        if (gi < N) {
          al = Sal[gi]; z = Sz[gi]; r = Sr[gi]; g = Sgc[gi]; bl = Sb[gi];
        }
        sAl[t] = al; sZ[t] = z; sR[t] = r; sG[t] = g; sB[t] = bl;
      }

      // ---- per-wave tile-intersection bitmask via ballot (SGPR, no LDS) ----
      int gi16 = b * 16 + col;
      int flag = 0;
      if (lane < 16 && gi16 < N) {
        int bnd = Sbnd[gi16];
        int t0x = bnd & 255, t1x = (bnd >> 8) & 255;
        int t0y = (bnd >> 16) & 255, t1y = (bnd >> 24) & 255;
        flag = Svld[gi16] && (tx >= t0x) && (tx <= t1x) && (ty >= t0y) && (ty <= t1y);
      }
      unsigned mask = (unsigned)__ballot(flag) & 0xFFFFu;

      if (mask) {  // wave-uniform; EXEC all ones inside
        bool gok = gi16 < N;
        float a1x = gok ? pA1x[gi16] : 0.0f;
        float a1y = gok ? pA1y[gi16] : 0.0f;
        float a2x = (gok && !half) ? Sg4[gi16] : 0.0f;
        float a2y = (gok && !half) ? Sg5[gi16] : 0.0f;
        v2f A1 = {a1x, a1y}, A2 = {a2x, a2y};

        v8f d0 = {};
        d0 = wmma_gs(A1, B1g0, d0);  // K = 0..3
        d0 = wmma_gs(A2, B2g0, d0);  // K = 4..5 (6,7 zero-padded)
        v8f d1 = {};
        d1 = wmma_gs(A1, B1g1, d1);
        d1 = wmma_gs(A2, B2g1, d1);
#pragma unroll
        for (int vi = 0; vi < 8; ++vi) sPow[vi * 256 + pbase0] = d0[vi];
#pragma unroll
        for (int vi = 0; vi < 8; ++vi) sPow[vi * 256 + pbase1] = d1[vi];
      }
      // barrier #2: sAl/sPow visible to whole block
      __syncthreads();

      if (mask) {
        for (int j = 0; j < 16; ++j) {
          if (!((mask >> j) & 1u)) continue;  // scalar bit-test, no LDS
          float power = sPow[j * 256 + t];
          float ai = fminf(sAl[j] * expf(fminf(power, 0.0f)), 0.99f);
          if (ai < GS_ALPHA_THR || T <= GS_T_THR) continue;
          float w = T * ai;
          rgb0 += w * sR[j]; rgb1 += w * sG[j]; rgb2 += w * sB[j];
          dep += w * sZ[j]; acc += w; cnt += 1;
          T *= (1.0f - ai);
        }
      }
    }

    int x = tx * GS_TILE + px_i, y = ty * GS_TILE + py_i;
    if (x < W && y < H) {
      int p = y * W + x;
      out[3 * p + 0] = rgb0;
      out[3 * p + 1] = rgb1;
      out[3 * p + 2] = rgb2;
      out[HW * 3 + p] = dep / fmaxf(acc, 1e-6f);
      out[HW * 3 + HW + p] = (float)cnt;
    }
  }
}

extern "C" void kernel_launch(void* const* d_in, const int* in_sizes, int n_in,
                              void* d_out, int out_size, void* d_ws, size_t ws_size,
                              hipStream_t stream) {
  const float* pc = (const float*)d_in[0];
  const float* feat = (const float*)d_in[1];
  // d_in[2] = point_object_id (unused)
  const int* inval = (const int*)d_in[3];
  const float* Km = (const float*)d_in[4];
  const float* qv = (const float*)d_in[5];
  const float* tv = (const float*)d_in[6];
  const int* Hp = (const int*)d_in[7];
  const int* Wp = (const int*)d_in[8];
  const int* bandp = (const int*)d_in[9];
  float* ws = (float*)d_ws;
  float* out = (float*)d_out;

  int N = in_sizes[0] / 3;
  int blocks = (N + 255) / 256;

  gs_preprocess<<<blocks, 256, 0, stream>>>(pc, feat, inval, Km, qv, tv, Hp, Wp,
                                            bandp, ws, N);
  gs_sort_scatter<<<blocks, 256, 0, stream>>>(ws, N);
  // H/W live on device; grid-stride over tiles with a fixed grid.
  gs_raster<<<128, 256, 0, stream>>>(ws, Hp, Wp, out, N);
}